// LlamaAttention_heavy_hitter_7584912245148
// MI455X (gfx1250) — compile-verified
//
#include <hip/hip_runtime.h>
#include <hip/hip_bf16.h>
#include <math.h>

#define B_    4
#define HID_  4096
#define H_    16
#define KVH_  4
#define D_    256
#define PAST_ 8191
#define S_    8192
#define KKEEP_ 2048
#define R_    128

// workspace layout (float offsets)
#define OFF_QROPE  0u          // B*H*D            = 16384
#define OFF_KROPE  16384u      // B*KVH*D          = 4096
#define OFF_VNEW   20480u      // B*KVH*D          = 4096
#define OFF_QMASK  24576u      // B*H*D            = 16384
#define OFF_RSCALE 40960u      // B*H              = 64
#define OFF_APPROX 41024u      // B*H*S            = 524288
#define OFF_EXACT  565312u     // B*H*S            = 524288
#define OFF_THR    1089600u    // B*H (uint)       = 64
#define OFF_MVAL   1089664u    // B*H              = 64
#define OFF_ZINV   1089728u    // B*H              = 64
#define OFF_YACC   1089792u    // B*H*D            = 16384
#define OFF_ZPAD   1106176u    // 256 zero floats (A-matrix rows 8..15), contiguous w/ YACC
// total = 1106432 floats (~4.43 MB)

typedef __attribute__((ext_vector_type(2))) float v2f;
typedef __attribute__((ext_vector_type(8))) float v8f;

__device__ __forceinline__ unsigned f2ord(float f) {
  unsigned u = __float_as_uint(f);
  return (u & 0x80000000u) ? ~u : (u | 0x80000000u);  // monotone float->uint
}

// ---------------- Kernel 1: QKV projection (GEMV, weight row shared over 4 batches) ----
__global__ void k_qkv(const float* __restrict__ hidden,
                      const float* __restrict__ Wq,
                      const float* __restrict__ Wk,
                      const float* __restrict__ Wv,
                      float* __restrict__ ws) {
  const int row = blockIdx.x, tid = threadIdx.x;
  const float* w;
  if (row < 4096)      w = Wq + (size_t)row * HID_;
  else if (row < 5120) w = Wk + (size_t)(row - 4096) * HID_;
  else                 w = Wv + (size_t)(row - 5120) * HID_;
  float a0 = 0.f, a1 = 0.f, a2 = 0.f, a3 = 0.f;
  for (int c = tid * 4; c < HID_; c += 1024) {
    __builtin_prefetch(w + c + 2048, 0, 0);
    float4 wv = *(const float4*)(w + c);
    float4 h0 = *(const float4*)(hidden + 0 * HID_ + c);
    float4 h1 = *(const float4*)(hidden + 1 * HID_ + c);
    float4 h2 = *(const float4*)(hidden + 2 * HID_ + c);
    float4 h3 = *(const float4*)(hidden + 3 * HID_ + c);
    a0 += wv.x * h0.x + wv.y * h0.y + wv.z * h0.z + wv.w * h0.w;
    a1 += wv.x * h1.x + wv.y * h1.y + wv.z * h1.z + wv.w * h1.w;
    a2 += wv.x * h2.x + wv.y * h2.y + wv.z * h2.z + wv.w * h2.w;
    a3 += wv.x * h3.x + wv.y * h3.y + wv.z * h3.z + wv.w * h3.w;
  }
  __shared__ float r0[256], r1[256], r2[256], r3[256];
  r0[tid] = a0; r1[tid] = a1; r2[tid] = a2; r3[tid] = a3;
  __syncthreads();
  for (int s = 128; s > 0; s >>= 1) {
    if (tid < s) { r0[tid] += r0[tid+s]; r1[tid] += r1[tid+s]; r2[tid] += r2[tid+s]; r3[tid] += r3[tid+s]; }
    __syncthreads();
  }
  if (tid == 0) {
    float vb[4] = { r0[0], r1[0], r2[0], r3[0] };
    for (int b = 0; b < B_; ++b) {
      if (row < 4096)      ws[OFF_QROPE + b * 4096 + row] = vb[b];
      else if (row < 5120) ws[OFF_KROPE + b * 1024 + (row - 4096)] = vb[b];
      else                 ws[OFF_VNEW  + b * 1024 + (row - 5120)] = vb[b];
    }
  }
}

// ---------------- Kernel 2: RoPE (in place, each thread owns a (t, t+128) pair) --------
__global__ void k_rope(float* __restrict__ ws, const int* __restrict__ pos_ids) {
  int slot = blockIdx.x;
  int b = slot / (H_ + KVH_), s = slot % (H_ + KVH_);
  float* x = (s < H_) ? (ws + OFF_QROPE + (size_t)(b * H_ + s) * D_)
                      : (ws + OFF_KROPE + (size_t)(b * KVH_ + (s - H_)) * D_);
  int t = threadIdx.x;  // 0..127
  float pos = (float)pos_ids[b];
  float inv = __expf(-((float)(2 * t) / (float)D_) * __logf(10000.0f));
  float sn, cs;
  __sincosf(pos * inv, &sn, &cs);
  float x0 = x[t], x1 = x[t + 128];
  x[t]       = x0 * cs - x1 * sn;   // q*cos + (-q_hi)*sin
  x[t + 128] = x1 * cs + x0 * sn;   // q*cos + ( q_lo)*sin
}

// ---------------- Kernel 3: per-(b,h) top-128 channel select + scale -------------------
__global__ void k_chan(float* __restrict__ ws) {
  int bh = blockIdx.x, tid = threadIdx.x;  // 256 threads, one per channel
  const float* q = ws + OFF_QROPE + (size_t)bh * D_;
  float qv = q[tid], a = fabsf(qv);
  __shared__ unsigned keys[256];
  __shared__ int cnt;
  __shared__ float red[256];
  unsigned key = f2ord(a);
  keys[tid] = key;
  __syncthreads();
  unsigned prefix = 0; int K = R_;            // radix-select 128th largest |q|
  for (int bit = 31; bit >= 0; --bit) {
    unsigned bm = 1u << bit;
    unsigned himask = (bit == 31) ? 0u : (0xFFFFFFFFu << (bit + 1));
    if (tid == 0) cnt = 0;
    __syncthreads();
    int pred = (((keys[tid] & himask) == prefix) && (keys[tid] & bm)) ? 1 : 0;
    atomicAdd(&cnt, pred);
    __syncthreads();
    int c = cnt;
    __syncthreads();
    if (c >= K) prefix |= bm; else K -= c;
  }
  int sel = (key >= prefix) ? 1 : 0;
  red[tid] = a; __syncthreads();
  for (int s = 128; s > 0; s >>= 1) { if (tid < s) red[tid] += red[tid+s]; __syncthreads(); }
  float s_all = red[0]; __syncthreads();
  red[tid] = sel ? a : 0.0f; __syncthreads();
  for (int s = 128; s > 0; s >>= 1) { if (tid < s) red[tid] += red[tid+s]; __syncthreads(); }
  float s_sel = red[0];
  ws[OFF_QMASK + (size_t)bh * D_ + tid] = sel ? qv : 0.0f;
  if (tid == 0) {
    float scale = sqrtf((float)D_ * s_sel / s_all);
    ws[OFF_RSCALE + bh] = 1.0f / scale;
  }
}

// ---------------- Kernel 4: WMMA f32 score pass (exact + approx in one sweep) ----------
// Per wave: C 16x16, A rows 0-3 = q(4 GQA heads), rows 4-7 = masked q, rows 8-15 read a
// zeroed pad region (no divergence). B = key^T tile (16 positions).
// fp32 exact: V_WMMA_F32_16X16X4_F32, K-loop of 64 unconditional b64 load pairs.
__global__ void k_scores(const float* __restrict__ past_key,
                         const float* __restrict__ attn_mask,
                         float* __restrict__ ws) {
  const int CH = S_ / 128;               // 64 chunks per (b,kvh); 8 waves x 16 positions
  int blk = blockIdx.x;
  int chunk = blk % CH, bkv = blk / CH;
  int b = bkv / KVH_, kvh = bkv % KVH_;
  int wave = threadIdx.x >> 5, lane = threadIdx.x & 31;
  int m = lane & 15, hi = lane >> 4;
  int pos0 = chunk * 128 + wave * 16;

  // A stream: lane holds row (lane&15) at K = k0 + 2*hi .. +1  (contiguous float2)
  const float* aptr;
  if (m < 4)      aptr = ws + OFF_QROPE + (size_t)(b * H_ + kvh * 4 + m) * D_ + 2 * hi;
  else if (m < 8) aptr = ws + OFF_QMASK + (size_t)(b * H_ + kvh * 4 + (m - 4)) * D_ + 2 * hi;
  else            aptr = ws + OFF_ZPAD + 2 * hi;          // zero rows: real loads of zeros

  // B stream: lane holds column (lane&15) -> key row pos0+(lane&15), same K pair
  int bpos = pos0 + m;
  const float* kr = (bpos < PAST_)
      ? past_key + ((size_t)(b * KVH_ + kvh) * PAST_ + bpos) * D_
      : ws + OFF_KROPE + (size_t)(b * KVH_ + kvh) * D_;   // new token's key
  const float* bptr = kr + 2 * hi;

  v8f c = { 0.f, 0.f, 0.f, 0.f, 0.f, 0.f, 0.f, 0.f };
#pragma unroll 8
  for (int k0 = 0; k0 < D_; k0 += 4) {
    v2f av = *(const v2f*)(aptr + k0);
    v2f bv = *(const v2f*)(bptr + k0);
    c = __builtin_amdgcn_wmma_f32_16x16x4_f32(false, av, false, bv, (short)0, c, false, false);
  }

  if (lane < 16) {                       // C rows 0-7 live in lanes 0-15 (N = lane)
    int p = pos0 + lane;
    float msk = attn_mask[(size_t)b * S_ + p];
    float exv[4] = { c[0], c[1], c[2], c[3] };
    float apv[4] = { c[4], c[5], c[6], c[7] };
#pragma unroll
    for (int g = 0; g < 4; ++g) {
      int h = kvh * 4 + g;
      ws[OFF_EXACT  + (size_t)(b * H_ + h) * S_ + p] = exv[g] * 0.0625f + msk;  // /sqrt(256)
      ws[OFF_APPROX + (size_t)(b * H_ + h) * S_ + p] = apv[g] * ws[OFF_RSCALE + b * H_ + h] + msk;
    }
  }
}

// ---------------- Kernel 5: radix-select the 2048th largest approx score --------------
__global__ void k_thresh(float* __restrict__ ws) {
  int bh = blockIdx.x, tid = threadIdx.x;   // 256 threads
  __shared__ unsigned keys[S_];             // 32 KB in LDS
  __shared__ int cnt;
  const float* ap = ws + OFF_APPROX + (size_t)bh * S_;
  for (int j = tid; j < S_; j += 256) keys[j] = f2ord(ap[j]);
  __syncthreads();
  unsigned prefix = 0; int K = KKEEP_;
  for (int bit = 31; bit >= 0; --bit) {
    unsigned bm = 1u << bit;
    unsigned himask = (bit == 31) ? 0u : (0xFFFFFFFFu << (bit + 1));
    if (tid == 0) cnt = 0;
    __syncthreads();
    int lc = 0;
    for (int j = tid; j < S_; j += 256) {
      unsigned k = keys[j];
      lc += (((k & himask) == prefix) && (k & bm)) ? 1 : 0;
    }
    atomicAdd(&cnt, lc);
    __syncthreads();
    int c = cnt;
    __syncthreads();
    if (c >= K) prefix |= bm; else K -= c;
  }
  if (tid == 0) ((unsigned*)ws)[OFF_THR + bh] = prefix;
}

// ---------------- Kernel 6: softmax stats (max & sum-exp over survivors) ---------------
__global__ void k_stats(float* __restrict__ ws) {
  int bh = blockIdx.x, tid = threadIdx.x;
  unsigned T = ((const unsigned*)ws)[OFF_THR + bh];
  const float* ap = ws + OFF_APPROX + (size_t)bh * S_;
  const float* ep = ws + OFF_EXACT + (size_t)bh * S_;
  __shared__ float red[256];
  float mloc = -3.402823466e38f;
  for (int j = tid; j < S_; j += 256)
    if (f2ord(ap[j]) >= T) mloc = fmaxf(mloc, ep[j]);
  red[tid] = mloc; __syncthreads();
  for (int s = 128; s > 0; s >>= 1) { if (tid < s) red[tid] = fmaxf(red[tid], red[tid+s]); __syncthreads(); }
  float mv = red[0]; __syncthreads();
  float zloc = 0.f;
  for (int j = tid; j < S_; j += 256)
    if (f2ord(ap[j]) >= T) zloc += __expf(ep[j] - mv);
  red[tid] = zloc; __syncthreads();
  for (int s = 128; s > 0; s >>= 1) { if (tid < s) red[tid] += red[tid+s]; __syncthreads(); }
  if (tid == 0) { ws[OFF_MVAL + bh] = mv; ws[OFF_ZINV + bh] = 1.0f / red[0]; }
}

// ---------------- Kernel 7: probs @ V (only ~2048 selected rows per head) --------------
__global__ void k_wv(const float* __restrict__ past_value, float* __restrict__ ws) {
  const int NCH = S_ / 512;                 // 16 position chunks
  int blk = blockIdx.x;
  int ch = blk % NCH, bh = blk / NCH;
  int b = bh / H_, h = bh % H_, kvh = h >> 2;
  int tid = threadIdx.x;                    // thread owns channel d = tid (coalesced V rows)
  unsigned T = ((const unsigned*)ws)[OFF_THR + bh];
  float mv = ws[OFF_MVAL + bh], zi = ws[OFF_ZINV + bh];
  const float* ap = ws + OFF_APPROX + (size_t)bh * S_;
  const float* ep = ws + OFF_EXACT + (size_t)bh * S_;
  float acc = 0.f;
  int j0 = ch * 512;
  for (int j = j0; j < j0 + 512; ++j) {
    if (f2ord(ap[j]) >= T) {                // uniform branch across block
      float w = __expf(ep[j] - mv) * zi;
      const float* vr = (j < PAST_)
          ? past_value + ((size_t)(b * KVH_ + kvh) * PAST_ + j) * D_
          : ws + OFF_VNEW + (size_t)(b * KVH_ + kvh) * D_;
      acc += w * vr[tid];
    }
  }
  atomicAdd(ws + OFF_YACC + (size_t)bh * D_ + tid, acc);
}

// ---------------- Kernel 8: output projection GEMV -------------------------------------
__global__ void k_oproj(const float* __restrict__ Wo,
                        const float* __restrict__ ws,
                        float* __restrict__ out) {
  int row = blockIdx.x, tid = threadIdx.x;
  const float* w = Wo + (size_t)row * HID_;
  const float* y = ws + OFF_YACC;           // [b][h*256+d] == flat [b][4096]
  float a0 = 0.f, a1 = 0.f, a2 = 0.f, a3 = 0.f;
  for (int c = tid * 4; c < HID_; c += 1024) {
    __builtin_prefetch(w + c + 2048, 0, 0);
    float4 wv = *(const float4*)(w + c);
    float4 y0 = *(const float4*)(y + 0 * HID_ + c);
    float4 y1 = *(const float4*)(y + 1 * HID_ + c);
    float4 y2 = *(const float4*)(y + 2 * HID_ + c);
    float4 y3 = *(const float4*)(y + 3 * HID_ + c);
    a0 += wv.x * y0.x + wv.y * y0.y + wv.z * y0.z + wv.w * y0.w;
    a1 += wv.x * y1.x + wv.y * y1.y + wv.z * y1.z + wv.w * y1.w;
    a2 += wv.x * y2.x + wv.y * y2.y + wv.z * y2.z + wv.w * y2.w;
    a3 += wv.x * y3.x + wv.y * y3.y + wv.z * y3.z + wv.w * y3.w;
  }
  __shared__ float r0[256], r1[256], r2[256], r3[256];
  r0[tid] = a0; r1[tid] = a1; r2[tid] = a2; r3[tid] = a3;
  __syncthreads();
  for (int s = 128; s > 0; s >>= 1) {
    if (tid < s) { r0[tid] += r0[tid+s]; r1[tid] += r1[tid+s]; r2[tid] += r2[tid+s]; r3[tid] += r3[tid+s]; }
    __syncthreads();
  }
  if (tid == 0) {
    out[0 * HID_ + row] = r0[0];
    out[1 * HID_ + row] = r1[0];
    out[2 * HID_ + row] = r2[0];
    out[3 * HID_ + row] = r3[0];
  }
}

extern "C" void kernel_launch(void* const* d_in, const int* in_sizes, int n_in,
                              void* d_out, int out_size, void* d_ws, size_t ws_size,
                              hipStream_t stream) {
  const float* hidden = (const float*)d_in[0];
  const float* amask  = (const float*)d_in[1];
  const int*   posids = (const int*)d_in[2];
  const float* pkey   = (const float*)d_in[3];
  const float* pval   = (const float*)d_in[4];
  const float* Wq     = (const float*)d_in[5];
  const float* Wk     = (const float*)d_in[6];
  const float* Wv     = (const float*)d_in[7];
  const float* Wo     = (const float*)d_in[8];
  float* ws  = (float*)d_ws;
  float* out = (float*)d_out;

  // zero YACC accumulator + the contiguous A-matrix zero-pad region in one memset
  hipMemsetAsync(ws + OFF_YACC, 0, (size_t)(B_ * H_ * D_ + 256) * sizeof(float), stream);

  k_qkv   <<<6144,                 256, 0, stream>>>(hidden, Wq, Wk, Wv, ws);
  k_rope  <<<B_ * (H_ + KVH_),     128, 0, stream>>>(ws, posids);
  k_chan  <<<B_ * H_,              256, 0, stream>>>(ws);
  k_scores<<<B_ * KVH_ * (S_/128), 256, 0, stream>>>(pkey, amask, ws);
  k_thresh<<<B_ * H_,              256, 0, stream>>>(ws);
  k_stats <<<B_ * H_,              256, 0, stream>>>(ws);
  k_wv    <<<B_ * H_ * (S_/512),   256, 0, stream>>>(pval, ws);
  k_oproj <<<4096,                 256, 0, stream>>>(Wo, ws, out);
}